// CGMMLayer_37864431682174
// MI455X (gfx1250) — compile-verified
//
#include <hip/hip_runtime.h>
#include <hip/hip_bf16.h>
#include <stdint.h>

// Problem constants (fixed by the reference file)
#define NN  50000
#define EE  400000
#define CC  10
#define LL  3
#define GG  16
#define MM  32
#define NGR 500
#define CLG 480               // C*L*G

// LDS / workspace table layout (floats): sm_Q | sm_B | sm_SP
#define TAB_Q    0            // C*C*L*G = 4800
#define TAB_B    4800         // C*M*G   = 5120
#define TAB_SP   9920         // L*G     = 48
#define TAB_SIZE 9968         // dwords (39,872 bytes)

// Workspace byte offsets
#define WS_TAB_B    0u
#define WS_ROWCNT_B 40960u
#define WS_FILL_B   (40960u + 204800u)
#define WS_ROWPTR_B (40960u + 2u * 204800u)
#define WS_CSR_B    (40960u + 3u * 204800u)

typedef unsigned int v4u __attribute__((ext_vector_type(4)));
typedef int          v8i __attribute__((ext_vector_type(8)));
typedef int          v4i __attribute__((ext_vector_type(4)));

// ---------------------------------------------------------------- init
__global__ void k_init(int* __restrict__ rowcnt, int* __restrict__ fill,
                       float* __restrict__ lik) {
  int i = blockIdx.x * blockDim.x + threadIdx.x;
  if (i < NN) { rowcnt[i] = 0; fill[i] = 0; }
  if (i < NGR * GG) lik[i] = 0.0f;
}

// ---------------------------------------------------------------- softmax tables
__global__ void k_tables(const float* __restrict__ Q, const float* __restrict__ B,
                         const float* __restrict__ SP, float* __restrict__ tab) {
  int t = threadIdx.x;
  // sm_Q: softmax over leading C dim; 480 independent columns, stride CLG
  for (int r = t; r < CLG; r += blockDim.x) {
    float mx = -3.4e38f;
    for (int c = 0; c < CC; ++c) mx = fmaxf(mx, Q[c * CLG + r]);
    float s = 0.0f;
    for (int c = 0; c < CC; ++c) s += expf(Q[c * CLG + r] - mx);
    float inv = 1.0f / s;
    for (int c = 0; c < CC; ++c)
      tab[TAB_Q + c * CLG + r] = expf(Q[c * CLG + r] - mx) * inv;
  }
  // sm_B: softmax over M dim; columns (c,g), stride G
  for (int r = t; r < CC * GG; r += blockDim.x) {
    int c = r >> 4, g = r & 15;
    int base = c * (MM * GG) + g;
    float mx = -3.4e38f;
    for (int m = 0; m < MM; ++m) mx = fmaxf(mx, B[base + m * GG]);
    float s = 0.0f;
    for (int m = 0; m < MM; ++m) s += expf(B[base + m * GG] - mx);
    float inv = 1.0f / s;
    for (int m = 0; m < MM; ++m)
      tab[TAB_B + base + m * GG] = expf(B[base + m * GG] - mx) * inv;
  }
  // sm_SP: softmax over L dim; 16 columns
  for (int g = t; g < GG; g += blockDim.x) {
    float mx = -3.4e38f;
    for (int l = 0; l < LL; ++l) mx = fmaxf(mx, SP[l * GG + g]);
    float s = 0.0f;
    for (int l = 0; l < LL; ++l) s += expf(SP[l * GG + g] - mx);
    float inv = 1.0f / s;
    for (int l = 0; l < LL; ++l)
      tab[TAB_SP + l * GG + g] = expf(SP[l * GG + g] - mx) * inv;
  }
}

// ---------------------------------------------------------------- in-degree
__global__ void k_degree(const int* __restrict__ dst, int* __restrict__ rowcnt) {
  int i = blockIdx.x * blockDim.x + threadIdx.x;
  if (i < EE) atomicAdd(&rowcnt[dst[i]], 1);
}

// ---------------------------------------------------------------- exclusive scan (1 block)
__global__ void k_scan(const int* __restrict__ cnt, int* __restrict__ rowptr) {
  __shared__ int sm[256];
  __shared__ int s_run;
  if (threadIdx.x == 0) s_run = 0;
  __syncthreads();
  for (int base = 0; base < NN; base += 256) {
    int i = base + threadIdx.x;
    int v = (i < NN) ? cnt[i] : 0;
    sm[threadIdx.x] = v;
    __syncthreads();
    for (int off = 1; off < 256; off <<= 1) {
      int t = (threadIdx.x >= off) ? sm[threadIdx.x - off] : 0;
      __syncthreads();
      sm[threadIdx.x] += t;
      __syncthreads();
    }
    int run = s_run;
    if (i < NN) rowptr[i] = run + sm[threadIdx.x] - v;
    __syncthreads();
    if (threadIdx.x == 255) s_run = run + sm[255];
    __syncthreads();
  }
  if (threadIdx.x == 0) rowptr[NN] = s_run;
}

// ---------------------------------------------------------------- CSR scatter
__global__ void k_scatter(const int* __restrict__ dst, const int* __restrict__ src,
                          int* __restrict__ fill, const int* __restrict__ rowptr,
                          int* __restrict__ csr) {
  int i = blockIdx.x * blockDim.x + threadIdx.x;
  if (i < EE) {
    int d = dst[i];
    int slot = atomicAdd(&fill[d], 1);
    csr[rowptr[d] + slot] = src[i];
  }
}

// ---------------------------------------------------------------- fused main kernel
__global__ void __launch_bounds__(256)
k_main(const int* __restrict__ x, const float* __restrict__ ph0,
       const int* __restrict__ ph1, const int* __restrict__ batch,
       const float* __restrict__ tab, const int* __restrict__ rowptr,
       const int* __restrict__ csr, float* __restrict__ out_lik,
       float* __restrict__ out_hval, float* __restrict__ out_hidx) {
  __shared__ float s_tab[TAB_SIZE];

  // --- Stage the 39,872-byte softmax tables into LDS with the Tensor Data Mover.
  // 1-D D# per CDNA5 ISA §8.3/8.4: count=1, type=2("image"), data_size=4B,
  // tensor_dim0 = tile_dim0 = TAB_SIZE dwords. Wave 0 issues; others wait at barrier.
  if (threadIdx.x < 32) {
    uint64_t ga = (uint64_t)(uintptr_t)tab;
    uint32_t lo = (uint32_t)(uintptr_t)(&s_tab[0]);
    v4u g0;
    g0[0] = 1u;                                               // count=1, user descriptor
    g0[1] = lo;                                               // lds_addr (bytes)
    g0[2] = (uint32_t)ga;                                     // global_addr[31:0]
    g0[3] = ((uint32_t)(ga >> 32) & 0x01FFFFFFu) | (2u << 30); // addr[56:32] | type=2
    v8i g1;
    g1[0] = (2 << 16);                                        // data_size = 4 bytes
    g1[1] = (int)((TAB_SIZE & 0xFFFF) << 16);                 // tensor_dim0[15:0]
    g1[2] = (TAB_SIZE >> 16) | (1 << 16);                     // dim0[31:16] | tensor_dim1=1
    g1[3] = (int)((TAB_SIZE & 0xFFFF) << 16);                 // tile_dim0
    g1[4] = 1;                                                // tile_dim1=1, tile_dim2=0
    g1[5] = TAB_SIZE;                                         // tensor_dim0_stride
    g1[6] = 0;
    g1[7] = 0;
    v4i gz; gz[0] = 0; gz[1] = 0; gz[2] = 0; gz[3] = 0;
    v8i gz8; gz8[0] = 0; gz8[1] = 0; gz8[2] = 0; gz8[3] = 0;
    gz8[4] = 0; gz8[5] = 0; gz8[6] = 0; gz8[7] = 0;
    // 6-arg form (amdgpu-toolchain clang-23 / therock-10.0 headers)
    __builtin_amdgcn_tensor_load_to_lds(g0, g1, gz, gz, gz8, 0);
    __builtin_amdgcn_s_wait_tensorcnt(0);
  }
  __syncthreads();

  const float* sQ  = s_tab + TAB_Q;
  const float* sB  = s_tab + TAB_B;
  const float* sSP = s_tab + TAB_SP;

  int g    = threadIdx.x & (GG - 1);       // one lane per gaussian g
  int node = blockIdx.x * 16 + (threadIdx.x >> 4);
  if (node >= NN) return;

  // segment-sum over in-edges, fully in registers: acc[c][l] for this g
  float acc[CC][LL];
#pragma unroll
  for (int c = 0; c < CC; ++c)
#pragma unroll
    for (int l = 0; l < LL; ++l) acc[c][l] = 0.0f;

  int rs = rowptr[node], re = rowptr[node + 1];
  for (int j = rs; j < re; ++j) {
    int src = csr[j];
    int sbase = src * (LL * GG) + g;
#pragma unroll
    for (int l = 0; l < LL; ++l) {
      int   h1 = ph1[sbase + l * GG];
      float h0 = ph0[sbase + l * GG];
      const float* q = sQ + (h1 * LL + l) * GG + g;   // sm_Q[c, h1, l, g], c-stride CLG
#pragma unroll
      for (int c = 0; c < CC; ++c) acc[c][l] = fmaf(q[c * CLG], h0, acc[c][l]);
    }
  }

  float icnt = 1.0f / fmaxf((float)(re - rs), 1.0f);
  int   xi   = x[node];

  float em[CC], lem[CC];
#pragma unroll
  for (int c = 0; c < CC; ++c) {
    em[c]  = sB[c * (MM * GG) + xi * GG + g];
    lem[c] = logf(em[c]);
  }
  float spv[LL], lsp[LL];
#pragma unroll
  for (int l = 0; l < LL; ++l) {
    float sp = sSP[l * GG + g];
    spv[l] = sp * icnt;            // sm_SP * (1/max(cnt,1)) folded together
    lsp[l] = logf(sp);
  }

  // posterior_unnorm and its (c,l) sum for this (node, g)
  float Z = 0.0f;
#pragma unroll
  for (int c = 0; c < CC; ++c)
#pragma unroll
    for (int l = 0; l < LL; ++l) {
      float pu = acc[c][l] * spv[l] * em[c];
      acc[c][l] = pu;
      Z += pu;
    }
  float invZ = 1.0f / (Z + 1e-12f);

  int nb = node * (LL * GG) + g;
  int h1i[LL];
#pragma unroll
  for (int l = 0; l < LL; ++l) h1i[l] = ph1[nb + l * GG];

  float lik = 0.0f;
  float plsum[LL] = {0.0f, 0.0f, 0.0f};
  float best = 0.0f; int bidx = 0;
#pragma unroll
  for (int c = 0; c < CC; ++c) {
    float qi = 0.0f;
#pragma unroll
    for (int l = 0; l < LL; ++l) {
      float p = acc[c][l] * invZ;
      qi += p;
      plsum[l] += p;
      lik += p * logf(sQ[c * CLG + (h1i[l] * LL + l) * GG + g]);  // log(trans_i)
    }
    lik += qi * lem[c];                                            // B_lhood term
    if (c == 0 || qi > best) { best = qi; bidx = c; }              // first-max tie-break
  }
#pragma unroll
  for (int l = 0; l < LL; ++l) lik += plsum[l] * lsp[l];           // SP_lhood term

  out_hval[node * GG + g] = best;
  out_hidx[node * GG + g] = (float)bidx;
  atomicAdd(&out_lik[batch[node] * GG + g], lik);
}

// ---------------------------------------------------------------- launch
extern "C" void kernel_launch(void* const* d_in, const int* in_sizes, int n_in,
                              void* d_out, int out_size, void* d_ws, size_t ws_size,
                              hipStream_t stream) {
  const int*   x     = (const int*)d_in[0];
  const float* ph0   = (const float*)d_in[1];
  const int*   ph1   = (const int*)d_in[2];
  const int*   ei    = (const int*)d_in[3];
  const int*   batch = (const int*)d_in[4];
  // d_in[5] = num_graphs scalar (fixed NGR=500)
  const float* Q     = (const float*)d_in[6];
  const float* B     = (const float*)d_in[7];
  const float* SP    = (const float*)d_in[8];

  float* out_lik  = (float*)d_out;                 // (NG, G)
  float* out_hval = out_lik + NGR * GG;            // (N, 1, G)
  float* out_hidx = out_hval + NN * GG;            // (N, G) as float

  char*  ws     = (char*)d_ws;
  float* tab    = (float*)(ws + WS_TAB_B);
  int*   rowcnt = (int*)(ws + WS_ROWCNT_B);
  int*   fill   = (int*)(ws + WS_FILL_B);
  int*   rowptr = (int*)(ws + WS_ROWPTR_B);
  int*   csrcol = (int*)(ws + WS_CSR_B);

  const int* dst = ei;        // edge_index[0]
  const int* src = ei + EE;   // edge_index[1]

  k_init<<<(NN + 255) / 256, 256, 0, stream>>>(rowcnt, fill, out_lik);
  k_tables<<<1, 512, 0, stream>>>(Q, B, SP, tab);
  k_degree<<<(EE + 255) / 256, 256, 0, stream>>>(dst, rowcnt);
  k_scan<<<1, 256, 0, stream>>>(rowcnt, rowptr);
  k_scatter<<<(EE + 255) / 256, 256, 0, stream>>>(dst, src, fill, rowptr, csrcol);
  k_main<<<(NN + 15) / 16, 256, 0, stream>>>(x, ph0, ph1, batch, tab, rowptr,
                                             csrcol, out_lik, out_hval, out_hidx);
}